// BilinearAttention2_74380243632829
// MI455X (gfx1250) — compile-verified
//
#include <hip/hip_runtime.h>
#include <cstddef>

// ---------------------------------------------------------------------------
// CDNA5 / gfx1250 bilinear graph attention
//   D=256, H=8, HD=32, G=16, NA=NB=128
// Kernel 1: x1 = A@W1^T + b1 ; x2 = B@W2^T + b2   via v_wmma_f32_16x16x4_f32
// Kernel 2: per-(g,h) block: async global->LDS tile fill, tanh-bilinear
//           row/col means (v_tanh_f32), softmax, weighted pooling.
// ---------------------------------------------------------------------------

typedef __attribute__((ext_vector_type(2))) float v2f;
typedef __attribute__((ext_vector_type(8))) float v8f;
typedef __attribute__((ext_vector_type(4))) int   v4i;

#if defined(__AMDGCN__) && \
    __has_builtin(__builtin_amdgcn_global_load_async_to_lds_b128) && \
    __has_builtin(__builtin_amdgcn_s_wait_asynccnt)
#define USE_ASYNC_LDS 1
typedef __attribute__((address_space(1))) v4i g_v4i;   // global (prints __device__)
typedef __attribute__((address_space(3))) v4i l_v4i;   // LDS    (prints __shared__)
#else
#define USE_ASYNC_LDS 0
#endif

__device__ __forceinline__ float fast_tanh(float x) {
#if __has_builtin(__builtin_amdgcn_tanhf)
  return __builtin_amdgcn_tanhf(x);            // v_tanh_f32 (TRANS pipe)
#elif __has_builtin(__builtin_amdgcn_tanh_f32)
  return __builtin_amdgcn_tanh_f32(x);
#else
  return tanhf(x);
#endif
}

// ---------------------------------------------------------------------------
// Kernel 1: fused dual GEMM + bias.  blockIdx.y selects (A,W1,b1)->x1 vs
// (B,W2,b2)->x2.  One wave computes one 16x16 output tile over K=256
// (64 chained v_wmma_f32_16x16x4_f32).  8 waves / 256-thread block.
// ---------------------------------------------------------------------------
__global__ __launch_bounds__(256) void gemm_bias_wmma(
    const float* __restrict__ A, const float* __restrict__ B,
    const float* __restrict__ W1, const float* __restrict__ W2,
    const float* __restrict__ b1, const float* __restrict__ b2,
    float* __restrict__ x1, float* __restrict__ x2) {
  const float* X    = (blockIdx.y == 0) ? A  : B;
  const float* W    = (blockIdx.y == 0) ? W1 : W2;
  const float* bias = (blockIdx.y == 0) ? b1 : b2;
  float*       out  = (blockIdx.y == 0) ? x1 : x2;

  const int lane = threadIdx.x & 31;
  const int wave = threadIdx.x >> 5;
  const int tile = blockIdx.x * 8 + wave;     // 0..2047 (128 M-tiles x 16 N-tiles)
  const int m0 = (tile >> 4) * 16;
  const int n0 = (tile & 15) * 16;

  const int lrow = lane & 15;                 // M (A-frag) / N (B-frag) index
  const int koff = (lane >> 4) * 2;           // K sub-offset per half-wave

  // out[m][n] = sum_e X[m][e] * W[n][e]  (X @ W^T) -> B-frag[k][n] = W[n][k]
  const float* ap = X + (size_t)(m0 + lrow) * 256 + koff;
  const float* bp = W + (size_t)(n0 + lrow) * 256 + koff;

  v8f acc = {};
#pragma unroll 8
  for (int k = 0; k < 256; k += 4) {
    v2f a = *(const v2f*)(ap + k);
    v2f b = *(const v2f*)(bp + k);
    acc = __builtin_amdgcn_wmma_f32_16x16x4_f32(
        /*neg_a=*/false, a, /*neg_b=*/false, b,
        /*c_mod=*/(short)0, acc, /*reuse_a=*/false, /*reuse_b=*/false);
  }

  const float bv = bias[n0 + lrow];           // bias depends on column only
  const int rbase = (lane >> 4) * 8;          // D rows: r + 8*(lane>=16)
#pragma unroll
  for (int r = 0; r < 8; ++r)
    out[(size_t)(m0 + rbase + r) * 256 + (n0 + lrow)] = acc[r] + bv;
}

// ---------------------------------------------------------------------------
// Kernel 2: one 256-thread workgroup per (g,h) pair (128 blocks).
// ---------------------------------------------------------------------------
__device__ __forceinline__ void softmax128(float* v, float* tmp, int t, float scale) {
  if (t < 128) { v[t] *= scale; tmp[t] = v[t]; }
  __syncthreads();
  for (int s = 64; s > 0; s >>= 1) {
    if (t < s) tmp[t] = fmaxf(tmp[t], tmp[t + s]);
    __syncthreads();
  }
  const float mx = tmp[0];
  __syncthreads();
  if (t < 128) { float e = __expf(v[t] - mx); v[t] = e; tmp[t] = e; }
  __syncthreads();
  for (int s = 64; s > 0; s >>= 1) {
    if (t < s) tmp[t] += tmp[t + s];
    __syncthreads();
  }
  const float sum = tmp[0];
  __syncthreads();
  if (t < 128) v[t] *= (1.0f / sum);
  __syncthreads();
}

__global__ __launch_bounds__(256) void bilinear_attn_pool(
    const float* __restrict__ x1, const float* __restrict__ x2,
    const float* __restrict__ q, float* __restrict__ out) {
  __shared__ float sx1[128 * 32];
  __shared__ float sx2[128 * 32];
  __shared__ float srow[128];
  __shared__ float scol[128];
  __shared__ float stmp[128];
  __shared__ float sq[32];

  const int t  = threadIdx.x;
  const int gh = blockIdx.x;                  // 0..127
  const int g  = gh >> 3;
  const int h  = gh & 7;

  // Contiguous 4096-float block for this (g,h) -- the torch view is a pure
  // C-order reinterpretation of the GEMM output.
  const float* p1f = x1 + (size_t)g * 32768 + (size_t)h * 4096;
  const float* p2f = x2 + (size_t)g * 32768 + (size_t)h * 4096;

#if USE_ASYNC_LDS
  // Direct global->LDS async copy (GLOBAL_LOAD_ASYNC_TO_LDS_B128): each lane
  // moves 16B per issue, 4 issues per buffer, tracked by ASYNCcnt.
#pragma unroll
  for (int idx = 0; idx < 4; ++idx) {
    const int e = (t + idx * 256) * 4;        // float index, 16B aligned
    __builtin_amdgcn_global_load_async_to_lds_b128(
        (g_v4i*)(p1f + e), (l_v4i*)&sx1[e], 0, 0);
    __builtin_amdgcn_global_load_async_to_lds_b128(
        (g_v4i*)(p2f + e), (l_v4i*)&sx2[e], 0, 0);
  }
#else
  {
    const float4* p1 = (const float4*)p1f;
    const float4* p2 = (const float4*)p2f;
    float4* d1 = (float4*)sx1;
    float4* d2 = (float4*)sx2;
#pragma unroll
    for (int idx = 0; idx < 4; ++idx) {       // 1024 float4 / 256 threads
      d1[t + idx * 256] = p1[t + idx * 256];
      d2[t + idx * 256] = p2[t + idx * 256];
    }
  }
#endif

  if (t < 128) { srow[t] = 0.0f; scol[t] = 0.0f; }
  if (t < 32)  { sq[t] = q[t]; }

#if USE_ASYNC_LDS
  __builtin_amdgcn_s_wait_asynccnt(0);        // s_wait_asynccnt 0
#endif
  __syncthreads();

  float qr[32];
#pragma unroll
  for (int k = 0; k < 32; ++k) qr[k] = sq[k];

  // 16x16 thread grid of 8x8 (i,j) tiles covering 128x128.
  const int i0 = (t & 15) * 8;
  const int j0 = (t >> 4) * 8;
  float racc[8] = {};

#pragma unroll 1
  for (int jj = 0; jj < 8; ++jj) {
    float x2r[32];
#pragma unroll
    for (int k = 0; k < 32; ++k) x2r[k] = sx2[(j0 + jj) * 32 + k];
    float csum = 0.0f;
#pragma unroll
    for (int ii = 0; ii < 8; ++ii) {
      const float* r1 = &sx1[(i0 + ii) * 32];
      float s = 0.0f;
#pragma unroll
      for (int k = 0; k < 32; ++k)
        s += fast_tanh(r1[k] * x2r[k]) * qr[k];   // 32 v_tanh_f32 + FMAs
      racc[ii] += s;
      csum += s;
    }
    atomicAdd(&scol[j0 + jj], csum);              // ds_add_f32
  }
#pragma unroll
  for (int ii = 0; ii < 8; ++ii) atomicAdd(&srow[i0 + ii], racc[ii]);
  __syncthreads();

  // means (1/128) + softmax over nodes, in place in LDS
  softmax128(srow, stmp, t, 1.0f / 128.0f);       // b2a over i
  softmax128(scol, stmp, t, 1.0f / 128.0f);       // a2b over j

  // Weighted pooling: A_p[k] = sum_i sx1[i][k]*srow[i] ; B_p likewise.
  if (t < 32) {
    float acc = 0.0f;
    for (int i = 0; i < 128; ++i) acc += sx1[i * 32 + t] * srow[i];
    out[(size_t)g * 512 + h * 32 + t] = acc;
  } else if (t < 64) {
    const int k = t - 32;
    float acc = 0.0f;
    for (int i = 0; i < 128; ++i) acc += sx2[i * 32 + k] * scol[i];
    out[(size_t)g * 512 + 256 + h * 32 + k] = acc;
  }
}

// ---------------------------------------------------------------------------
extern "C" void kernel_launch(void* const* d_in, const int* in_sizes, int n_in,
                              void* d_out, int out_size, void* d_ws, size_t ws_size,
                              hipStream_t stream) {
  (void)in_sizes; (void)n_in; (void)out_size; (void)ws_size;
  const float* A  = (const float*)d_in[0];
  /* d_in[1] = batch1 (unused: uniform graphs) */
  const float* B  = (const float*)d_in[2];
  /* d_in[3] = batch2 (unused) */
  const float* W1 = (const float*)d_in[4];
  const float* b1 = (const float*)d_in[5];
  const float* W2 = (const float*)d_in[6];
  const float* b2 = (const float*)d_in[7];
  const float* q  = (const float*)d_in[8];
  float* out = (float*)d_out;

  float* x1 = (float*)d_ws;                    // [2048, 256] fp32
  float* x2 = x1 + (size_t)2048 * 256;         // [2048, 256] fp32 (4 MB total)

  dim3 grid1(256, 2);                          // 2048 tiles x {x1,x2}
  gemm_bias_wmma<<<grid1, 256, 0, stream>>>(A, B, W1, W2, b1, b2, x1, x2);
  bilinear_attn_pool<<<128, 256, 0, stream>>>(x1, x2, q, out);
}